// SimpleMemoryBlock_7610682049118
// MI455X (gfx1250) — compile-verified
//
#include <hip/hip_runtime.h>
#include <hip/hip_bf16.h>
#include <math.h>

// Problem constants (fixed by the reference)
#define BN 8192   // batch
#define DN 1024   // feature dim
#define KN 8192   // num codes
#define SN 64     // code dim
#define H1 128    // 2*S
#define H2 2048   // 2*D
#define EPSV 1e-5f

typedef float v2f __attribute__((ext_vector_type(2)));
typedef float v8f __attribute__((ext_vector_type(8)));

// D = A(16x4 f32) * B(4x16 f32) + C(16x16 f32), exact f32 on the matrix pipe.
__device__ __forceinline__ v8f wmma_f32(v2f a, v2f b, v8f c) {
  return __builtin_amdgcn_wmma_f32_16x16x4_f32(false, a, false, b, (short)0, c,
                                               false, false);
}

// ---- fragment loaders per ISA 7.12.2 layouts (wave32) ----
// A 16x4: lanes 0-15 rows, v.x = K(k0+2*half), v.y = K(k0+2*half+1)
__device__ __forceinline__ v2f load_a_rm(const float* M, int ld, int m0, int k0,
                                         int lane) {
  int half = lane >> 4;
  int r = m0 + (lane & 15);
  int k = k0 + half * 2;
  v2f a;
  a.x = M[(size_t)r * ld + k];
  a.y = M[(size_t)r * ld + k + 1];
  return a;
}
// B 4x16 from row-major W[ld cols]: lanes index columns, half selects K pair
__device__ __forceinline__ v2f load_b_rm(const float* W, int ld, int k0, int n0,
                                         int lane) {
  int half = lane >> 4;
  int c = n0 + (lane & 15);
  int k = k0 + half * 2;
  v2f b;
  b.x = W[(size_t)k * ld + c];
  b.y = W[(size_t)(k + 1) * ld + c];
  return b;
}
// B 4x16 where columns are codebook rows (codebook used transposed)
__device__ __forceinline__ v2f load_b_code(const float* C, int j0, int k0,
                                           int lane) {
  int half = lane >> 4;
  int code = j0 + (lane & 15);
  int k = k0 + half * 2;
  v2f b;
  b.x = C[(size_t)code * SN + k];
  b.y = C[(size_t)code * SN + k + 1];
  return b;
}

#define LDS_STRIDE 130  // 128 + pad to break bank conflicts

// ===================== Kernel 1: encode =====================
// h = relu(X@W1+b1); h = LN(h)*g+b; z = h@W2+b2
__global__ __launch_bounds__(128) void encode_kernel(
    const float* __restrict__ X, const float* __restrict__ W1,
    const float* __restrict__ B1, const float* __restrict__ G1,
    const float* __restrict__ Bt1, const float* __restrict__ W2,
    const float* __restrict__ B2, float* __restrict__ Z) {
  __shared__ float h[64 * LDS_STRIDE];
  const int wave = threadIdx.x >> 5;
  const int lane = threadIdx.x & 31;
  const int half = lane >> 4;
  const int ln = lane & 15;
  const int m0 = blockIdx.x * 64 + wave * 16;

  const v8f zero8 = {0.f, 0.f, 0.f, 0.f, 0.f, 0.f, 0.f, 0.f};
  v8f acc[8];
#pragma unroll
  for (int t = 0; t < 8; ++t) acc[t] = zero8;

  // GEMM1: [16 x 1024] x [1024 x 128]
  for (int k0 = 0; k0 < DN; k0 += 4) {
    v2f a = load_a_rm(X, DN, m0, k0, lane);
#pragma unroll
    for (int t = 0; t < 8; ++t) {
      v2f b = load_b_rm(W1, H1, k0, t * 16, lane);
      acc[t] = wmma_f32(a, b, acc[t]);
    }
  }
  // bias + relu -> LDS
#pragma unroll
  for (int t = 0; t < 8; ++t) {
    int col = t * 16 + ln;
    float bias = B1[col];
#pragma unroll
    for (int i = 0; i < 8; ++i) {
      int lr = wave * 16 + i + half * 8;
      float v = acc[t][i] + bias;
      h[lr * LDS_STRIDE + col] = fmaxf(v, 0.f);
    }
  }
  __syncthreads();
  // LayerNorm over 128 cols; one thread per row (64 rows)
  if (threadIdx.x < 64) {
    int row = threadIdx.x;
    float s = 0.f, s2 = 0.f;
    for (int c = 0; c < H1; ++c) {
      float v = h[row * LDS_STRIDE + c];
      s += v;
      s2 += v * v;
    }
    float mu = s * (1.f / H1);
    float var = s2 * (1.f / H1) - mu * mu;
    float rstd = rsqrtf(var + EPSV);
    for (int c = 0; c < H1; ++c) {
      float v = h[row * LDS_STRIDE + c];
      h[row * LDS_STRIDE + c] = (v - mu) * rstd * G1[c] + Bt1[c];
    }
  }
  __syncthreads();
  // GEMM2: [16 x 128] x [128 x 64] from LDS
  v8f acc2[4];
#pragma unroll
  for (int t = 0; t < 4; ++t) acc2[t] = zero8;
  for (int k0 = 0; k0 < H1; k0 += 4) {
    int lr = wave * 16 + ln;
    int k = k0 + half * 2;
    v2f a;
    a.x = h[lr * LDS_STRIDE + k];
    a.y = h[lr * LDS_STRIDE + k + 1];
#pragma unroll
    for (int t = 0; t < 4; ++t) {
      v2f b = load_b_rm(W2, SN, k0, t * 16, lane);
      acc2[t] = wmma_f32(a, b, acc2[t]);
    }
  }
#pragma unroll
  for (int t = 0; t < 4; ++t) {
    int col = t * 16 + ln;
    float bias = B2[col];
#pragma unroll
    for (int i = 0; i < 8; ++i) {
      int row = m0 + i + half * 8;
      Z[(size_t)row * SN + col] = acc2[t][i] + bias;
    }
  }
}

// ===================== Kernel 2a: codebook norms =====================
__global__ __launch_bounds__(256) void cnorm_kernel(const float* __restrict__ C,
                                                    float* __restrict__ cn) {
  int k = blockIdx.x * 256 + threadIdx.x;
  if (k < KN) {
    float s = 0.f;
#pragma unroll 8
    for (int i = 0; i < SN; ++i) {
      float v = C[(size_t)k * SN + i];
      s += v * v;
    }
    cn[k] = s;
  }
}

// ===================== Kernel 2b: fused distance GEMM + argmin ============
__global__ __launch_bounds__(128) void argmin_kernel(
    const float* __restrict__ Z, const float* __restrict__ C,
    const float* __restrict__ cn, int* __restrict__ idx) {
  const int wave = threadIdx.x >> 5;
  const int lane = threadIdx.x & 31;
  const int half = lane >> 4;
  const int m0 = blockIdx.x * 64 + wave * 16;
  const v8f zero8 = {0.f, 0.f, 0.f, 0.f, 0.f, 0.f, 0.f, 0.f};

  // preload the full z A-tile (16x64): 16 k-steps of 4
  v2f a[16];
#pragma unroll
  for (int kk = 0; kk < 16; ++kk) a[kk] = load_a_rm(Z, SN, m0, kk * 4, lane);

  float bestv[8];
  int besti[8];
#pragma unroll
  for (int i = 0; i < 8; ++i) {
    bestv[i] = INFINITY;
    besti[i] = 0;
  }

  for (int j0 = 0; j0 < KN; j0 += 16) {
    v8f acc = zero8;
#pragma unroll
    for (int kk = 0; kk < 16; ++kk) {
      v2f b = load_b_code(C, j0, kk * 4, lane);
      acc = wmma_f32(a[kk], b, acc);
    }
    int code = j0 + (lane & 15);
    float cv = cn[code];
#pragma unroll
    for (int i = 0; i < 8; ++i) {
      float score = cv - 2.f * acc[i];  // ||z||^2 constant per row: drop it
      if (score < bestv[i]) {
        bestv[i] = score;
        besti[i] = code;
      }
    }
  }
  // reduce (min, first-index) across the 16 lanes of each half-wave
#pragma unroll
  for (int off = 1; off < 16; off <<= 1) {
#pragma unroll
    for (int i = 0; i < 8; ++i) {
      float ov = __shfl_xor(bestv[i], off, 16);
      int oi = __shfl_xor(besti[i], off, 16);
      if (ov < bestv[i] || (ov == bestv[i] && oi < besti[i])) {
        bestv[i] = ov;
        besti[i] = oi;
      }
    }
  }
  if ((lane & 15) == 0) {
#pragma unroll
    for (int i = 0; i < 8; ++i) idx[m0 + i + half * 8] = besti[i];
  }
}

// ===================== Kernel 3: decode stage 1 =====================
// r_pre = relu(codebook[idx]@rec_w1 + b1); store r_pre + per-row mu/rstd
__global__ __launch_bounds__(128) void decode1_kernel(
    const float* __restrict__ C, const int* __restrict__ idx,
    const float* __restrict__ W1, const float* __restrict__ B1,
    float* __restrict__ Rpre, float* __restrict__ Mu,
    float* __restrict__ Rstd) {
  const int wave = threadIdx.x >> 5;
  const int lane = threadIdx.x & 31;
  const int half = lane >> 4;
  const int ln = lane & 15;
  const int m0 = blockIdx.x * 64 + wave * 16;
  const v8f zero8 = {0.f, 0.f, 0.f, 0.f, 0.f, 0.f, 0.f, 0.f};

  // gather A-tile: row r uses codebook[idx[r]]
  int r = m0 + ln;
  int code = idx[r];
  const float* src = C + (size_t)code * SN + half * 2;
  v2f a[16];
#pragma unroll
  for (int kk = 0; kk < 16; ++kk) {
    a[kk].x = src[kk * 4];
    a[kk].y = src[kk * 4 + 1];
  }

  float s[8], s2[8];
#pragma unroll
  for (int i = 0; i < 8; ++i) {
    s[i] = 0.f;
    s2[i] = 0.f;
  }

  for (int t = 0; t < H2 / 16; ++t) {  // 128 N-tiles
    v8f acc = zero8;
#pragma unroll
    for (int kk = 0; kk < 16; ++kk) {
      v2f b = load_b_rm(W1, H2, kk * 4, t * 16, lane);
      acc = wmma_f32(a[kk], b, acc);
    }
    int col = t * 16 + ln;
    float bias = B1[col];
#pragma unroll
    for (int i = 0; i < 8; ++i) {
      float v = fmaxf(acc[i] + bias, 0.f);
      s[i] += v;
      s2[i] += v * v;
      int row = m0 + i + half * 8;
      Rpre[(size_t)row * H2 + col] = v;
    }
  }
  // cross-lane sums (each half-wave owns its 8 rows)
#pragma unroll
  for (int off = 1; off < 16; off <<= 1) {
#pragma unroll
    for (int i = 0; i < 8; ++i) {
      s[i] += __shfl_xor(s[i], off, 16);
      s2[i] += __shfl_xor(s2[i], off, 16);
    }
  }
  if ((lane & 15) == 0) {
#pragma unroll
    for (int i = 0; i < 8; ++i) {
      int row = m0 + i + half * 8;
      float mu = s[i] * (1.f / H2);
      float var = s2[i] * (1.f / H2) - mu * mu;
      Mu[row] = mu;
      Rstd[row] = rsqrtf(var + EPSV);
    }
  }
}

// ===================== Kernel 4: decode stage 2 =====================
// out = LN(r_pre)*g+b @ rec_w2 + b2   (LN applied on the fly to A-frags)
__global__ __launch_bounds__(128) void decode2_kernel(
    const float* __restrict__ Rpre, const float* __restrict__ Mu,
    const float* __restrict__ Rstd, const float* __restrict__ G,
    const float* __restrict__ Bt, const float* __restrict__ W2,
    const float* __restrict__ B2, float* __restrict__ Out) {
  const int wave = threadIdx.x >> 5;
  const int lane = threadIdx.x & 31;
  const int half = lane >> 4;
  const int ln = lane & 15;
  const int m0 = blockIdx.x * 64 + wave * 16;
  const int n_base = blockIdx.y * 64;
  const v8f zero8 = {0.f, 0.f, 0.f, 0.f, 0.f, 0.f, 0.f, 0.f};

  int r = m0 + ln;
  float mu = Mu[r];
  float rstd = Rstd[r];
  const float* rrow = Rpre + (size_t)r * H2;

  v8f acc[4];
#pragma unroll
  for (int t = 0; t < 4; ++t) acc[t] = zero8;

  for (int k0 = 0; k0 < H2; k0 += 4) {
    int k = k0 + half * 2;
    v2f a;
    a.x = (rrow[k] - mu) * rstd * G[k] + Bt[k];
    a.y = (rrow[k + 1] - mu) * rstd * G[k + 1] + Bt[k + 1];
#pragma unroll
    for (int t = 0; t < 4; ++t) {
      v2f b = load_b_rm(W2, DN, k0, n_base + t * 16, lane);
      acc[t] = wmma_f32(a, b, acc[t]);
    }
  }
#pragma unroll
  for (int t = 0; t < 4; ++t) {
    int col = n_base + t * 16 + ln;
    float bias = B2[col];
#pragma unroll
    for (int i = 0; i < 8; ++i) {
      int row = m0 + i + half * 8;
      Out[(size_t)row * DN + col] = acc[t][i] + bias;
    }
  }
}

// ===================== finalize: usage + indices-as-float ================
__global__ __launch_bounds__(256) void zero_usage_kernel(float* __restrict__ u) {
  int k = blockIdx.x * 256 + threadIdx.x;
  if (k < KN) u[k] = 0.f;
}
__global__ __launch_bounds__(256) void scatter_kernel(
    const int* __restrict__ idx, float* __restrict__ out_idx,
    float* __restrict__ usage) {
  int b = blockIdx.x * 256 + threadIdx.x;
  if (b < BN) {
    int c = idx[b];
    out_idx[b] = (float)c;
    atomicAdd(&usage[c], 1.f);
  }
}

extern "C" void kernel_launch(void* const* d_in, const int* in_sizes, int n_in,
                              void* d_out, int out_size, void* d_ws,
                              size_t ws_size, hipStream_t stream) {
  (void)in_sizes; (void)n_in; (void)out_size; (void)ws_size;
  const float* features = (const float*)d_in[0];
  const float* codebook = (const float*)d_in[1];
  const float* proj_w1  = (const float*)d_in[2];
  const float* proj_b1  = (const float*)d_in[3];
  const float* ln1_g    = (const float*)d_in[4];
  const float* ln1_b    = (const float*)d_in[5];
  const float* proj_w2  = (const float*)d_in[6];
  const float* proj_b2  = (const float*)d_in[7];
  const float* rec_w1   = (const float*)d_in[8];
  const float* rec_b1   = (const float*)d_in[9];
  const float* ln2_g    = (const float*)d_in[10];
  const float* ln2_b    = (const float*)d_in[11];
  const float* rec_w2   = (const float*)d_in[12];
  const float* rec_b2   = (const float*)d_in[13];

  // output layout: reconstructed [B*D] | indices [B] | usage [K], all f32
  float* out_rec = (float*)d_out;
  float* out_idx = out_rec + (size_t)BN * DN;
  float* out_usage = out_idx + BN;

  // workspace layout (~69.3 MB)
  float* ws = (float*)d_ws;
  float* Z    = ws;                       // B*S
  float* cn   = Z + (size_t)BN * SN;      // K
  int*   idx  = (int*)(cn + KN);          // B
  float* Mu   = (float*)(idx + BN);       // B
  float* Rstd = Mu + BN;                  // B
  float* Rpre = Rstd + BN;                // B*2D

  encode_kernel<<<BN / 64, 128, 0, stream>>>(features, proj_w1, proj_b1, ln1_g,
                                             ln1_b, proj_w2, proj_b2, Z);
  cnorm_kernel<<<KN / 256, 256, 0, stream>>>(codebook, cn);
  argmin_kernel<<<BN / 64, 128, 0, stream>>>(Z, codebook, cn, idx);
  decode1_kernel<<<BN / 64, 128, 0, stream>>>(codebook, idx, rec_w1, rec_b1,
                                              Rpre, Mu, Rstd);
  dim3 g2(BN / 64, DN / 64);
  decode2_kernel<<<g2, 128, 0, stream>>>(Rpre, Mu, Rstd, ln2_g, ln2_b, rec_w2,
                                         rec_b2, out_rec);
  zero_usage_kernel<<<KN / 256, 256, 0, stream>>>(out_usage);
  scatter_kernel<<<BN / 256, 256, 0, stream>>>(idx, out_idx, out_usage);
}